// BaseBiasedSVDpp_80925773791743
// MI455X (gfx1250) — compile-verified
//
#include <hip/hip_runtime.h>
#include <hip/hip_bf16.h>

typedef __attribute__((ext_vector_type(2))) float v2f;
typedef __attribute__((ext_vector_type(8))) float v8f;

#define FACTORS        64
#define HIST_LEN       50
#define ROWS_PER_BLOCK 16
#define LDS_STRIDE     66   // even padded stride: keeps float2 8B-aligned, kills bank conflicts

__global__ __launch_bounds__(512) void svdpp_pred_kernel(
    const int*   __restrict__ x,          // [batch, 2] int32
    const int*   __restrict__ items_hist, // [N_USERS, 50] int32
    const float* __restrict__ P,          // [N_USERS, 64]
    const float* __restrict__ Q,          // [N_ITEMS, 64]
    const float* __restrict__ bu,         // [N_USERS]
    const float* __restrict__ bi,         // [N_ITEMS]
    const float* __restrict__ Y,          // [N_ITEMS, 64]
    const float* __restrict__ inv_sqrt,   // [N_USERS]
    const float* __restrict__ mu_p,       // [1]
    float*       __restrict__ out,        // [batch]
    int batch)
{
    __shared__ float A_lds[ROWS_PER_BLOCK][LDS_STRIDE]; // pu_tilde rows (16 x 64)
    __shared__ float B_lds[ROWS_PER_BLOCK][LDS_STRIDE]; // qi rows       (16 x 64)
    __shared__ float C_lds[8][32];                      // WMMA C dump
    __shared__ float bias_lds[ROWS_PER_BLOCK];

    const int lane = threadIdx.x & 31;
    const int wave = threadIdx.x >> 5;
    const int e    = blockIdx.x * ROWS_PER_BLOCK + wave;

    // ---------------- Phase 1: per-wave gather + y_sum reduction --------------
    if (e < batch) {
        const int user = x[2 * e + 0];
        const int item = x[2 * e + 1];

        // Preload 50 history indices into 2 regs/lane, broadcast later via readlane.
        const int* hrow = items_hist + (long)user * HIST_LEN;
        const int  h0   = hrow[(lane < HIST_LEN) ? lane : 0];
        const int  i1   = lane + 32;
        const int  h1   = hrow[(i1 < HIST_LEN) ? i1 : 0];

        // Each lane owns factors {2*lane, 2*lane+1}.
        v2f acc0 = {0.f, 0.f};
        v2f acc1 = {0.f, 0.f};

        #pragma unroll 10
        for (int h = 0; h < HIST_LEN; ++h) {
            // wave-uniform index -> SGPR base address for the Y row gather
            const int idx = (h < 32) ? __builtin_amdgcn_readlane(h0, h)
                                     : __builtin_amdgcn_readlane(h1, h - 32);
            const v2f yv = *(const v2f*)(Y + (long)idx * FACTORS + 2 * lane);
            if (h & 1) acc1 += yv; else acc0 += yv;
        }
        const v2f ysum = acc0 + acc1;

        const float is = inv_sqrt[user];
        const v2f pu   = *(const v2f*)(P + (long)user * FACTORS + 2 * lane);
        const v2f qi   = *(const v2f*)(Q + (long)item * FACTORS + 2 * lane);
        const v2f pt   = pu + is * ysum;

        *(v2f*)&A_lds[wave][2 * lane] = pt;
        *(v2f*)&B_lds[wave][2 * lane] = qi;
        if (lane == 0)
            bias_lds[wave] = mu_p[0] + bu[user] + bi[item];
    }
    __syncthreads();

    // ---------------- Phase 2: wave 0 computes 16x16 = A(16x64) * B^T via WMMA ----
    // D[m][n] = sum_k pu_tilde[m][k] * qi[n][k]; we need the diagonal m==n.
    if (wave == 0) {
        v8f c = {};
        const int row  = lane & 15;            // A row M / B column N for this lane
        const int koff = (lane < 16) ? 0 : 2;  // lanes 16-31 hold K+2 per ISA layout

        #pragma unroll
        for (int kb = 0; kb < FACTORS / 4; ++kb) {
            const int k = 4 * kb + koff;
            v2f a, b;
            // A 16x4 f32 layout: VGPR0 = K=0 (lanes0-15) / K=2 (lanes16-31), VGPR1 = K=1 / K=3
            a.x = A_lds[row][k];
            a.y = A_lds[row][k + 1];
            // B 4x16 f32 layout mirrors C row striping: VGPR0 = K row 0 / 2, VGPR1 = K row 1 / 3
            b.x = B_lds[row][k];
            b.y = B_lds[row][k + 1];
            c = __builtin_amdgcn_wmma_f32_16x16x4_f32(
                    /*neg_a=*/false, a, /*neg_b=*/false, b,
                    /*c_mod=*/(short)0, c, /*reuse_a=*/false, /*reuse_b=*/false);
        }

        #pragma unroll
        for (int r = 0; r < 8; ++r)
            C_lds[r][lane] = c[r];
    }
    __syncthreads();

    // ---------------- Phase 3: extract diagonal, add biases, store -----------
    if (threadIdx.x < ROWS_PER_BLOCK) {
        const int m  = threadIdx.x;
        const int eo = blockIdx.x * ROWS_PER_BLOCK + m;
        if (eo < batch) {
            // C layout: VGPR r, lanes0-15 -> (M=r, N=lane); lanes16-31 -> (M=8+r, N=lane-16)
            const float d = C_lds[m & 7][(m < 8) ? m : (m + 16)];
            out[eo] = bias_lds[m] + d;
        }
    }
}

extern "C" void kernel_launch(void* const* d_in, const int* in_sizes, int n_in,
                              void* d_out, int out_size, void* d_ws, size_t ws_size,
                              hipStream_t stream) {
    const int*   x          = (const int*)  d_in[0];
    const int*   items_hist = (const int*)  d_in[1];
    const float* P          = (const float*)d_in[2];
    const float* Q          = (const float*)d_in[3];
    const float* bu         = (const float*)d_in[4];
    const float* bi         = (const float*)d_in[5];
    const float* Y          = (const float*)d_in[6];
    const float* inv_sqrt   = (const float*)d_in[7];
    const float* mu         = (const float*)d_in[8];
    float*       out        = (float*)d_out;

    const int batch  = in_sizes[0] / 2;
    const int blocks = (batch + ROWS_PER_BLOCK - 1) / ROWS_PER_BLOCK;
    svdpp_pred_kernel<<<blocks, 512, 0, stream>>>(
        x, items_hist, P, Q, bu, bi, Y, inv_sqrt, mu, out, batch);
}